// SSMBlock_21449066676464
// MI455X (gfx1250) — compile-verified
//
#include <hip/hip_runtime.h>
#include <math.h>

// ---------------- problem constants (match reference) ----------------
#define DMODEL 1024
#define EEXP   2048
#define NSTATE 16
#define GROUPS 16
#define GSIZE  128
#define KCONV  7
#define DFFN   4096
#define BSZ    2
#define TLEN   2048
#define MROWS  (BSZ * TLEN)   // 4096

typedef __attribute__((ext_vector_type(2))) float v2f;
typedef __attribute__((ext_vector_type(8))) float v8f;

// ---------------------------------------------------------------------
// Async global->LDS 16-byte copy (CDNA5 GLOBAL_LOAD_ASYNC_TO_LDS_B128,
// tracked by ASYNCcnt). LDS byte address is the low 32 bits of the
// generic pointer (ISA 10.2 aperture layout).
// ---------------------------------------------------------------------
__device__ __forceinline__ void async_copy_f4(void* lds_dst, const float* gsrc) {
  unsigned lds = (unsigned)(unsigned long long)lds_dst;
  asm volatile("global_load_async_to_lds_b128 %0, %1, off"
               :: "v"(lds), "v"(gsrc) : "memory");
}
__device__ __forceinline__ void wait_async0() {
  asm volatile("s_wait_asynccnt 0" ::: "memory");
}

// ---------------------------------------------------------------------
// WMMA fp32 GEMM:  C[M,N] = A[M,K] * B[K,N]   (row-major, fp32)
// Block tile 64x128, 8 waves (4 in M x 2 in N), each wave 16x64 output
// via four 16x16 accumulators. K-chunks of 16 -> 16 v_wmma per chunk
// per wave. LDS double-buffered; tiles staged with async-to-LDS and
// overlapped with WMMA compute.
// Requires: M % 64 == 0, K % 16 == 0, N % 16 == 0 (guarded tails).
// ---------------------------------------------------------------------
#define BM 64
#define BN 128
#define BK 16
#define APAD 4
#define BPAD 4

__global__ __launch_bounds__(256) void gemm_wmma_f32(
    const float* __restrict__ A, const float* __restrict__ Bw,
    float* __restrict__ C, int M, int Kd, int Nn)
{
  __shared__ float As[2][BM][BK + APAD];
  __shared__ float Bs[2][BK][BN + BPAD];

  const int tid   = threadIdx.x;
  const int wave  = tid >> 5;        // 0..7
  const int lane  = tid & 31;
  const int wm    = wave >> 1;       // 0..3  (M direction)
  const int wn    = wave & 1;        // 0..1  (N direction)
  const int l15   = lane & 15;
  const int khalf = lane >> 4;       // 0 or 1

  const int block_m = blockIdx.y * BM;
  const int block_n = blockIdx.x * BN;

  // A staging: 64x16 = 1024 floats = 256 threads x 1 float4
  const int a_row = tid >> 2;          // 0..63
  const int a_col = (tid & 3) * 4;     // 0,4,8,12
  // B staging: 16x128 = 2048 floats = 256 threads x 2 float4
  const int b_row = tid >> 4;          // 0..15
  const int b_col = (tid & 15) * 8;    // 0..120
  const int gcol0 = block_n + b_col;
  const int gcol1 = gcol0 + 4;
  const bool b_in0 = (gcol0 < Nn);     // Nn % 16 == 0 -> float4-granular
  const bool b_in1 = (gcol1 < Nn);

  // zero-fill OOB B columns once (content is k-independent: always 0)
  if (!b_in0) {
    #pragma unroll
    for (int q = 0; q < 4; ++q) { Bs[0][b_row][b_col + q] = 0.f; Bs[1][b_row][b_col + q] = 0.f; }
  }
  if (!b_in1) {
    #pragma unroll
    for (int q = 4; q < 8; ++q) { Bs[0][b_row][b_col + q] = 0.f; Bs[1][b_row][b_col + q] = 0.f; }
  }

  v8f acc0 = {}, acc1 = {}, acc2 = {}, acc3 = {};

  // prologue: stage tile 0 into buffer 0
  {
    const float* srcA = A + (size_t)(block_m + a_row) * Kd + a_col;
    async_copy_f4(&As[0][a_row][a_col], srcA);
    const float* srcB = Bw + (size_t)b_row * Nn;
    if (b_in0) async_copy_f4(&Bs[0][b_row][b_col],     srcB + gcol0);
    if (b_in1) async_copy_f4(&Bs[0][b_row][b_col + 4], srcB + gcol1);
  }
  wait_async0();
  __syncthreads();

  int cur = 0;
  for (int k0 = 0; k0 < Kd; k0 += BK) {
    const int nxt = cur ^ 1;
    // stage next tile while computing on current one
    if (k0 + BK < Kd) {
      const float* srcA = A + (size_t)(block_m + a_row) * Kd + (k0 + BK + a_col);
      async_copy_f4(&As[nxt][a_row][a_col], srcA);
      const float* srcB = Bw + (size_t)(k0 + BK + b_row) * Nn;
      if (b_in0) async_copy_f4(&Bs[nxt][b_row][b_col],     srcB + gcol0);
      if (b_in1) async_copy_f4(&Bs[nxt][b_row][b_col + 4], srcB + gcol1);
    }

    // 4 K-steps of 4; fragment layouts per CDNA5 ISA 7.12.2
    #pragma unroll
    for (int kk = 0; kk < BK; kk += 4) {
      const int kk2 = kk + 2 * khalf;
      v2f a, b0, b1, b2, b3;
      a.x = As[cur][wm * 16 + l15][kk2 + 0];
      a.y = As[cur][wm * 16 + l15][kk2 + 1];
      const float* brow0 = &Bs[cur][kk2 + 0][wn * 64 + l15];
      const float* brow1 = &Bs[cur][kk2 + 1][wn * 64 + l15];
      b0.x = brow0[0];  b0.y = brow1[0];
      b1.x = brow0[16]; b1.y = brow1[16];
      b2.x = brow0[32]; b2.y = brow1[32];
      b3.x = brow0[48]; b3.y = brow1[48];
      acc0 = __builtin_amdgcn_wmma_f32_16x16x4_f32(false, a, false, b0, (short)0, acc0, false, false);
      acc1 = __builtin_amdgcn_wmma_f32_16x16x4_f32(false, a, false, b1, (short)0, acc1, false, false);
      acc2 = __builtin_amdgcn_wmma_f32_16x16x4_f32(false, a, false, b2, (short)0, acc2, false, false);
      acc3 = __builtin_amdgcn_wmma_f32_16x16x4_f32(false, a, false, b3, (short)0, acc3, false, false);
    }

    wait_async0();     // own async transfers complete
    __syncthreads();   // everyone's transfers + reads of 'cur' complete
    cur = nxt;
  }

  // store: acc vgpr r -> M = r (lanes 0-15) or r+8 (lanes 16-31), N = l15
  #pragma unroll
  for (int r = 0; r < 8; ++r) {
    const int row   = block_m + wm * 16 + r + 8 * khalf;
    const int cbase = block_n + wn * 64 + l15;
    if (cbase      < Nn) C[(size_t)row * Nn + cbase     ] = acc0[r];
    if (cbase + 16 < Nn) C[(size_t)row * Nn + cbase + 16] = acc1[r];
    if (cbase + 32 < Nn) C[(size_t)row * Nn + cbase + 32] = acc2[r];
    if (cbase + 48 < Nn) C[(size_t)row * Nn + cbase + 48] = acc3[r];
  }
}

// ---------------------------------------------------------------------
// RMSNorm: out[row] = x[row] * rsqrt(mean(x^2)+eps) * w
// ---------------------------------------------------------------------
__global__ __launch_bounds__(256) void rmsnorm_kernel(
    const float* __restrict__ x, const float* __restrict__ w,
    float* __restrict__ out, int Dd)
{
  const int row = blockIdx.x;
  const float* xr = x + (size_t)row * Dd;
  float ss = 0.f;
  for (int i = threadIdx.x; i < Dd; i += 256) { float v = xr[i]; ss += v * v; }
  #pragma unroll
  for (int off = 16; off > 0; off >>= 1) ss += __shfl_down(ss, off, 32);
  __shared__ float red[8];
  __shared__ float scale_s;
  if ((threadIdx.x & 31) == 0) red[threadIdx.x >> 5] = ss;
  __syncthreads();
  if (threadIdx.x == 0) {
    float tot = 0.f;
    #pragma unroll
    for (int i = 0; i < 8; ++i) tot += red[i];
    scale_s = rsqrtf(tot / (float)Dd + 1e-6f);
  }
  __syncthreads();
  const float sc = scale_s;
  for (int i = threadIdx.x; i < Dd; i += 256)
    out[(size_t)row * Dd + i] = xr[i] * sc * w[i];
}

// ---------------------------------------------------------------------
// Depthwise causal conv over the v-half of uv: v[b,t,e] =
//   sum_{k=0..6} conv_w[e,k] * uv[b, t+k-6, E + e]   (zero left pad)
// ---------------------------------------------------------------------
__global__ __launch_bounds__(256) void dwconv_kernel(
    const float* __restrict__ uv, const float* __restrict__ cw,
    float* __restrict__ v)
{
  const int idx = blockIdx.x * 256 + threadIdx.x;
  if (idx >= MROWS * EEXP) return;
  const int e = idx % EEXP;
  const int t = (idx / EEXP) % TLEN;
  const int b = idx / (EEXP * TLEN);
  float acc = 0.f;
  #pragma unroll
  for (int k = 0; k < KCONV; ++k) {
    const int ts = t + k - (KCONV - 1);
    if (ts >= 0)
      acc += cw[e * KCONV + k] *
             uv[((size_t)(b * TLEN + ts)) * (2 * EEXP) + EEXP + e];
  }
  v[idx] = acc;
}

// ---------------------------------------------------------------------
// Sequential selective scan. One wave per (b,g); lanes map to n = lane&15.
//   state = exp(dt*A)*state + dt*Bt ; y = sum_n state*Ct
// B/C/dt are staged 8 timesteps at a time into registers with a
// double-buffered prefetch so the recurrence never stalls on HBM latency.
// ---------------------------------------------------------------------
#define SCH 8
__global__ __launch_bounds__(32) void scan_kernel(
    const float* __restrict__ Bt, const float* __restrict__ Ct,
    const float* __restrict__ dtraw, const float* __restrict__ dt_b,
    const float* __restrict__ A_log, float* __restrict__ y)
{
  const int b = blockIdx.x >> 4;
  const int g = blockIdx.x & 15;
  const int lane = threadIdx.x;
  const int n = lane & 15;
  const float An = -expf(A_log[g * NSTATE + n]);
  const float bias = dt_b[g];
  float state = 0.f;

  const size_t base = (size_t)b * TLEN;
  const float* Bp = Bt    + (size_t)g * NSTATE + n;  // +row*256
  const float* Cp = Ct    + (size_t)g * NSTATE + n;  // +row*256
  const float* Dp = dtraw + g;                       // +row*16

  float Bc[SCH], Cc[SCH], Dc[SCH];
  float Bn_[SCH], Cn_[SCH], Dn_[SCH];

  #pragma unroll
  for (int j = 0; j < SCH; ++j) {
    const size_t row = base + j;
    Bc[j] = Bp[row * (GROUPS * NSTATE)];
    Cc[j] = Cp[row * (GROUPS * NSTATE)];
    Dc[j] = Dp[row * GROUPS];
  }

  for (int t0 = 0; t0 < TLEN; t0 += SCH) {
    if (t0 + SCH < TLEN) {
      #pragma unroll
      for (int j = 0; j < SCH; ++j) {
        const size_t row = base + t0 + SCH + j;
        Bn_[j] = Bp[row * (GROUPS * NSTATE)];
        Cn_[j] = Cp[row * (GROUPS * NSTATE)];
        Dn_[j] = Dp[row * GROUPS];
      }
    }
    #pragma unroll
    for (int j = 0; j < SCH; ++j) {
      float dtv = Dc[j] + bias;
      dtv = (dtv > 20.f) ? dtv : log1pf(expf(dtv));   // softplus
      state = expf(dtv * An) * state + dtv * Bc[j];
      float prod = state * Cc[j];
      prod += __shfl_xor(prod, 1, 32);
      prod += __shfl_xor(prod, 2, 32);
      prod += __shfl_xor(prod, 4, 32);
      prod += __shfl_xor(prod, 8, 32);
      if (lane == 0) y[(base + t0 + j) * GROUPS + g] = prod;
    }
    #pragma unroll
    for (int j = 0; j < SCH; ++j) { Bc[j] = Bn_[j]; Cc[j] = Cn_[j]; Dc[j] = Dn_[j]; }
  }
}

// ---------------------------------------------------------------------
// Gate: ygate[row,e] = sigmoid(u[row,e]) * y[row, e/128]
// ---------------------------------------------------------------------
__global__ __launch_bounds__(256) void gate_kernel(
    const float* __restrict__ uv, const float* __restrict__ y,
    float* __restrict__ ygate)
{
  const int idx = blockIdx.x * 256 + threadIdx.x;
  if (idx >= MROWS * EEXP) return;
  const int e = idx % EEXP;
  const int row = idx / EEXP;
  const float u = uv[(size_t)row * (2 * EEXP) + e];
  const float sig = 1.f / (1.f + expf(-u));
  ygate[idx] = sig * y[(size_t)row * GROUPS + (e >> 7)];
}

__global__ __launch_bounds__(256) void add_kernel(
    const float* __restrict__ a, const float* __restrict__ b,
    float* __restrict__ out, int n)
{
  const int idx = blockIdx.x * 256 + threadIdx.x;
  if (idx < n) out[idx] = a[idx] + b[idx];
}

__global__ __launch_bounds__(256) void swiglu_kernel(
    float* __restrict__ a1, const float* __restrict__ a3, int n)
{
  const int idx = blockIdx.x * 256 + threadIdx.x;
  if (idx < n) {
    const float v = a1[idx];
    const float sig = 1.f / (1.f + expf(-v));
    a1[idx] = v * sig * a3[idx];
  }
}

// ---------------------------------------------------------------------
extern "C" void kernel_launch(void* const* d_in, const int* in_sizes, int n_in,
                              void* d_out, int out_size, void* d_ws, size_t ws_size,
                              hipStream_t stream) {
  (void)in_sizes; (void)n_in; (void)out_size; (void)ws_size;

  const float* x          = (const float*)d_in[0];
  const float* norm1_w    = (const float*)d_in[1];
  const float* in_proj_w  = (const float*)d_in[2];
  const float* conv_w     = (const float*)d_in[3];
  const float* B_proj_w   = (const float*)d_in[4];
  const float* C_proj_w   = (const float*)d_in[5];
  const float* dt_proj_w  = (const float*)d_in[6];
  const float* dt_proj_b  = (const float*)d_in[7];
  const float* A_log      = (const float*)d_in[8];
  const float* out_proj_w = (const float*)d_in[9];
  const float* norm2_w    = (const float*)d_in[10];
  const float* w1         = (const float*)d_in[11];
  const float* w2         = (const float*)d_in[12];
  const float* w3         = (const float*)d_in[13];
  float* out = (float*)d_out;

  // workspace layout (floats)
  float* ws = (float*)d_ws;
  float* buf_h    = ws;                                   // 4096*1024
  float* buf_uv   = buf_h    + (size_t)MROWS * DMODEL;    // 4096*4096 (also a1)
  float* buf_v    = buf_uv   + (size_t)MROWS * 2 * EEXP;  // 4096*2048 (also ygate)
  float* buf_Bt   = buf_v    + (size_t)MROWS * EEXP;      // 4096*256
  float* buf_Ct   = buf_Bt   + (size_t)MROWS * GROUPS * NSTATE;
  float* buf_dt   = buf_Ct   + (size_t)MROWS * GROUPS * NSTATE;  // 4096*16
  float* buf_y    = buf_dt   + (size_t)MROWS * GROUPS;           // 4096*16
  float* buf_proj = buf_y    + (size_t)MROWS * GROUPS;    // 4096*1024
  float* buf_x2   = buf_proj + (size_t)MROWS * DMODEL;    // 4096*1024
  float* buf_a3   = buf_x2   + (size_t)MROWS * DMODEL;    // 4096*4096

  const int nBT  = MROWS;          // 4096 rows
  const int nBTE = MROWS * EEXP;
  const int nBTD = MROWS * DMODEL;

  // 1. h = rmsnorm(x) * norm1_w
  rmsnorm_kernel<<<nBT, 256, 0, stream>>>(x, norm1_w, buf_h, DMODEL);

  // 2. uv = h @ in_proj  [4096,1024]x[1024,4096]
  gemm_wmma_f32<<<dim3((2 * EEXP) / BN, nBT / BM), 256, 0, stream>>>(
      buf_h, in_proj_w, buf_uv, nBT, DMODEL, 2 * EEXP);

  // 3. v = causal depthwise conv of uv[:, E:2E]
  dwconv_kernel<<<(nBTE + 255) / 256, 256, 0, stream>>>(buf_uv, conv_w, buf_v);

  // 4. Bt / Ct / dt projections from v
  gemm_wmma_f32<<<dim3((GROUPS * NSTATE + BN - 1) / BN, nBT / BM), 256, 0, stream>>>(
      buf_v, B_proj_w, buf_Bt, nBT, EEXP, GROUPS * NSTATE);
  gemm_wmma_f32<<<dim3((GROUPS * NSTATE + BN - 1) / BN, nBT / BM), 256, 0, stream>>>(
      buf_v, C_proj_w, buf_Ct, nBT, EEXP, GROUPS * NSTATE);
  gemm_wmma_f32<<<dim3((GROUPS + BN - 1) / BN, nBT / BM), 256, 0, stream>>>(
      buf_v, dt_proj_w, buf_dt, nBT, EEXP, GROUPS);

  // 5. sequential scan -> y[B*T, G]
  scan_kernel<<<BSZ * GROUPS, 32, 0, stream>>>(
      buf_Bt, buf_Ct, buf_dt, dt_proj_b, A_log, buf_y);

  // 6. ygate = sigmoid(u) * repeat(y)   (overwrites buf_v)
  gate_kernel<<<(nBTE + 255) / 256, 256, 0, stream>>>(buf_uv, buf_y, buf_v);

  // 7. proj = ygate @ out_proj  [4096,2048]x[2048,1024]
  gemm_wmma_f32<<<dim3(DMODEL / BN, nBT / BM), 256, 0, stream>>>(
      buf_v, out_proj_w, buf_proj, nBT, EEXP, DMODEL);

  // 8. x2 = x + proj
  add_kernel<<<(nBTD + 255) / 256, 256, 0, stream>>>(x, buf_proj, buf_x2, nBTD);

  // 9. h2 = rmsnorm(x2) * norm2_w   (reuse buf_h)
  rmsnorm_kernel<<<nBT, 256, 0, stream>>>(buf_x2, norm2_w, buf_h, DMODEL);

  // 10. a1 = h2 @ w1 (into buf_uv) ; a3 = h2 @ w3
  gemm_wmma_f32<<<dim3(DFFN / BN, nBT / BM), 256, 0, stream>>>(
      buf_h, w1, buf_uv, nBT, DMODEL, DFFN);
  gemm_wmma_f32<<<dim3(DFFN / BN, nBT / BM), 256, 0, stream>>>(
      buf_h, w3, buf_a3, nBT, DMODEL, DFFN);

  // 11. a1 = silu(a1) * a3
  swiglu_kernel<<<(nBT * DFFN + 255) / 256, 256, 0, stream>>>(
      buf_uv, buf_a3, nBT * DFFN);

  // 12. ff = a1 @ w2  [4096,4096]x[4096,1024]  (into buf_h)
  gemm_wmma_f32<<<dim3(DMODEL / BN, nBT / BM), 256, 0, stream>>>(
      buf_uv, w2, buf_h, nBT, DFFN, DMODEL);

  // 13. out = x2 + ff
  add_kernel<<<(nBTD + 255) / 256, 256, 0, stream>>>(buf_x2, buf_h, out, nBTD);
}